// GCN_82068235092421
// MI455X (gfx1250) — compile-verified
//
#include <hip/hip_runtime.h>
#include <hip/hip_bf16.h>

// GCN forward, fused for MI455X (gfx1250, wave32).
// Bandwidth-bound (~570 MB total traffic @ 23.3 TB/s): WMMA f32 16x16x4 used
// for both dense GEMMs to minimize instruction overhead; SpMM gathers hit the
// 192 MB L2 (h1=6.4MB, o1=16MB resident).

#define NNODES  100000
#define NEDGES  3200000
#define INF     512
#define HIDF    16
#define OUTF    40
#define NTILES  (NNODES / 16)   // 6250, exact

typedef __attribute__((ext_vector_type(2))) float v2f;
typedef __attribute__((ext_vector_type(8))) float v8f;

// ---------------------------------------------------------------- row_ptr ---
// rp[i] = lower_bound(edge_row, i); rp[N] = E. edge_row is sorted.
__global__ void gcn_rowptr(const int* __restrict__ edge_row, int* __restrict__ rp) {
    int i = blockIdx.x * blockDim.x + threadIdx.x;
    if (i > NNODES) return;
    int lo = 0, hi = NEDGES;
    while (lo < hi) {
        int mid = (lo + hi) >> 1;
        if (edge_row[mid] < i) lo = mid + 1; else hi = mid;
    }
    rp[i] = lo;
}

// ------------------------------------------------- dropout1 + X @ W1 -------
// One wave per 16-row tile of x. A = sparse-dropout(x) (16x4 per step),
// B = w1 slice from LDS, C accumulates over K=512 in 128 WMMA steps.
__global__ void __launch_bounds__(256)
gcn_gemm1(const float* __restrict__ x, const float* __restrict__ drop1,
          const float* __restrict__ w1, float* __restrict__ h1) {
    __shared__ float lw1[INF * HIDF];                 // 32 KB
    for (int i = threadIdx.x; i < INF * HIDF; i += blockDim.x) lw1[i] = w1[i];
    __syncthreads();

    const int wave = threadIdx.x >> 5;                // 0..7
    const int lane = threadIdx.x & 31;
    const int tile = blockIdx.x * 8 + wave;           // wave-uniform
    if (tile >= NTILES) return;                       // EXEC stays all-1s

    const int m     = lane & 15;                      // row-in-tile (A) / col (B,D)
    const int khalf = (lane >> 4) << 1;               // 0 or 2
    const long row  = (long)tile * 16 + m;
    const float2* xr = (const float2*)(x + row * INF);
    const float2* dr = (const float2*)(drop1 + row * INF);

    v8f c = {0.f, 0.f, 0.f, 0.f, 0.f, 0.f, 0.f, 0.f};
    #pragma unroll 4
    for (int kk = 0; kk < INF; kk += 4) {
        const int k0 = kk + khalf;                    // even
        float2 xv = xr[k0 >> 1];
        float2 uv = dr[k0 >> 1];
        v2f a, b;
        a.x = (uv.x < 0.5f) ? xv.x * 2.0f : 0.0f;     // retain=0.5, scale=2
        a.y = (uv.y < 0.5f) ? xv.y * 2.0f : 0.0f;
        b.x = lw1[k0 * HIDF + m];
        b.y = lw1[(k0 + 1) * HIDF + m];
        c = __builtin_amdgcn_wmma_f32_16x16x4_f32(false, a, false, b,
                                                  (short)0, c, false, false);
    }
    // D layout: VGPR v -> row v (lanes 0-15) / row v+8 (lanes 16-31), col = m
    float* out = h1 + (long)tile * 16 * HIDF;
    const int rofs = (lane >> 4) << 3;
    #pragma unroll
    for (int v = 0; v < 8; ++v)
        out[(rofs + v) * HIDF + m] = c[v];
}

// ------------------------------------------- SpMM1 + bias + ReLU ----------
// Half-wave (16 lanes = HID features) per node; serial edge loop; 64B
// coalesced gathers from L2-resident h1.
__global__ void __launch_bounds__(256)
gcn_spmm1(const int* __restrict__ rp, const int* __restrict__ edge_col,
          const float* __restrict__ edge_val, const float* __restrict__ h1,
          const float* __restrict__ b1, float* __restrict__ h2) {
    const int node = blockIdx.x * 16 + (threadIdx.x >> 4);
    const int f    = threadIdx.x & 15;
    if (node >= NNODES) return;
    const int e0 = rp[node], e1 = rp[node + 1];
    float acc = 0.0f;
    for (int e = e0; e < e1; ++e) {
        const float v = edge_val[e];
        const long  cidx = edge_col[e];
        acc = fmaf(v, h1[cidx * HIDF + f], acc);
    }
    const float r = acc + b1[f];
    h2[(long)node * HIDF + f] = r > 0.0f ? r : 0.0f;
}

// ------------------------------------------------ dropout2 + H @ W2 -------
// K=16 (4 WMMA steps), N=40 -> 3 tiles of 16 (last zero-padded in LDS).
__global__ void __launch_bounds__(256)
gcn_gemm2(const float* __restrict__ h2, const float* __restrict__ drop2,
          const float* __restrict__ w2, float* __restrict__ o1) {
    __shared__ float lw2[HIDF * 48];                  // 16x48, padded
    for (int i = threadIdx.x; i < HIDF * 48; i += blockDim.x) {
        const int k = i / 48, n = i % 48;
        lw2[i] = (n < OUTF) ? w2[k * OUTF + n] : 0.0f;
    }
    __syncthreads();

    const int wave = threadIdx.x >> 5;
    const int lane = threadIdx.x & 31;
    const int tile = blockIdx.x * 8 + wave;           // wave-uniform
    if (tile >= NTILES) return;

    const int m     = lane & 15;
    const int khalf = (lane >> 4) << 1;
    const long row  = (long)tile * 16 + m;
    const float2* hr = (const float2*)(h2 + row * HIDF);
    const float2* dr = (const float2*)(drop2 + row * HIDF);

    v8f c0 = {0.f,0.f,0.f,0.f,0.f,0.f,0.f,0.f};
    v8f c1 = c0, c2 = c0;
    #pragma unroll
    for (int kk = 0; kk < HIDF; kk += 4) {
        const int k0 = kk + khalf;
        float2 xv = hr[k0 >> 1];
        float2 uv = dr[k0 >> 1];
        v2f a;
        a.x = (uv.x < 0.5f) ? xv.x * 2.0f : 0.0f;     // keep=0.5, scale=2
        a.y = (uv.y < 0.5f) ? xv.y * 2.0f : 0.0f;
        v2f b0, b1v, b2v;
        b0.x  = lw2[k0 * 48 + m];       b0.y  = lw2[(k0 + 1) * 48 + m];
        b1v.x = lw2[k0 * 48 + 16 + m];  b1v.y = lw2[(k0 + 1) * 48 + 16 + m];
        b2v.x = lw2[k0 * 48 + 32 + m];  b2v.y = lw2[(k0 + 1) * 48 + 32 + m];
        c0 = __builtin_amdgcn_wmma_f32_16x16x4_f32(false, a, false, b0,
                                                   (short)0, c0, false, false);
        c1 = __builtin_amdgcn_wmma_f32_16x16x4_f32(false, a, false, b1v,
                                                   (short)0, c1, false, false);
        c2 = __builtin_amdgcn_wmma_f32_16x16x4_f32(false, a, false, b2v,
                                                   (short)0, c2, false, false);
    }
    float* out = o1 + (long)tile * 16 * OUTF;
    const int rofs = (lane >> 4) << 3;
    #pragma unroll
    for (int v = 0; v < 8; ++v) {
        out[(rofs + v) * OUTF + m]      = c0[v];
        out[(rofs + v) * OUTF + 16 + m] = c1[v];
        if (m < 8)                                    // cols 32..39 only
            out[(rofs + v) * OUTF + 32 + m] = c2[v];
    }
}

// ------------------------------------------------- SpMM2 + bias -----------
// One wave per node; lanes cover 40 features as 32 + 8.
__global__ void __launch_bounds__(256)
gcn_spmm2(const int* __restrict__ rp, const int* __restrict__ edge_col,
          const float* __restrict__ edge_val, const float* __restrict__ o1,
          const float* __restrict__ b2, float* __restrict__ out) {
    const int wave = threadIdx.x >> 5;
    const int lane = threadIdx.x & 31;
    const int node = blockIdx.x * 8 + wave;
    if (node >= NNODES) return;
    const int e0 = rp[node], e1 = rp[node + 1];
    float acc0 = 0.0f, acc1 = 0.0f;
    for (int e = e0; e < e1; ++e) {
        const float v = edge_val[e];
        const long  cidx = edge_col[e];
        const float* orow = o1 + cidx * OUTF;
        acc0 = fmaf(v, orow[lane], acc0);
        if (lane < 8) acc1 = fmaf(v, orow[32 + lane], acc1);
    }
    out[(long)node * OUTF + lane] = acc0 + b2[lane];
    if (lane < 8)
        out[(long)node * OUTF + 32 + lane] = acc1 + b2[32 + lane];
}

// ---------------------------------------------------------------------------
extern "C" void kernel_launch(void* const* d_in, const int* in_sizes, int n_in,
                              void* d_out, int out_size, void* d_ws, size_t ws_size,
                              hipStream_t stream) {
    // setup_inputs() order:
    const float* x        = (const float*)d_in[0];
    const float* w1       = (const float*)d_in[1];
    const float* b1       = (const float*)d_in[2];
    const float* w2       = (const float*)d_in[3];
    const float* b2       = (const float*)d_in[4];
    const float* edge_val = (const float*)d_in[5];
    const float* drop1_u  = (const float*)d_in[6];
    const float* drop2_u  = (const float*)d_in[7];
    const int*   edge_row = (const int*)d_in[8];
    const int*   edge_col = (const int*)d_in[9];
    float* out = (float*)d_out;

    // workspace layout (512B-aligned chunks)
    char* ws = (char*)d_ws;
    size_t off = 0;
    int* rp = (int*)(ws + off);
    off += ((size_t)(NNODES + 1) * sizeof(int) + 511) & ~(size_t)511;
    float* h1 = (float*)(ws + off);
    off += ((size_t)NNODES * HIDF * sizeof(float) + 511) & ~(size_t)511;
    float* h2 = (float*)(ws + off);
    off += ((size_t)NNODES * HIDF * sizeof(float) + 511) & ~(size_t)511;
    float* o1 = (float*)(ws + off);

    gcn_rowptr<<<(NNODES + 256) / 256, 256, 0, stream>>>(edge_row, rp);
    gcn_gemm1 <<<(NTILES + 7) / 8, 256, 0, stream>>>(x, drop1_u, w1, h1);
    gcn_spmm1 <<<NNODES / 16, 256, 0, stream>>>(rp, edge_col, edge_val, h1, b1, h2);
    gcn_gemm2 <<<(NTILES + 7) / 8, 256, 0, stream>>>(h2, drop2_u, w2, o1);
    gcn_spmm2 <<<NNODES / 8, 256, 0, stream>>>(rp, edge_col, edge_val, o1, b2, out);
}